// _integrate_depths_88450556494132
// MI455X (gfx1250) — compile-verified
//
#include <hip/hip_runtime.h>
#include <hip/hip_bf16.h>

typedef float v2f __attribute__((ext_vector_type(2)));
typedef float v8f __attribute__((ext_vector_type(8)));

#define OBS_WEIGHT 1.0f
#define TH_BBOX 0.025f
#define TRUNC_MULT 3.0f
#define TSDF_TH 8.0f

#ifndef __has_builtin
#define __has_builtin(x) 0
#endif
#if __has_builtin(__builtin_amdgcn_wmma_f32_16x16x4_f32)
#define USE_WMMA 1
#endif

// ---------- float <-> order-preserving uint encoding for atomic min/max ----------
__device__ __forceinline__ unsigned encf(float f) {
  unsigned u = __float_as_uint(f);
  return (u & 0x80000000u) ? ~u : (u | 0x80000000u);
}
__device__ __forceinline__ float decf(unsigned e) {
  unsigned u = (e & 0x80000000u) ? (e & 0x7fffffffu) : ~e;
  return __uint_as_float(u);
}

// ---------- workspace init ----------
__global__ void k_init_ws(unsigned* bbox) {
  int t = threadIdx.x;
  if (t < 3) bbox[t] = 0xffffffffu;       // encoded +inf for mins
  else if (t < 6) bbox[t] = 0u;           // encoded -inf for maxs
}

// ---------- pass 1: world-space bbox over all valid depth pixels ----------
__global__ void k_bbox_reduce(const float* __restrict__ depths,
                              const int* __restrict__ masks,
                              const float* __restrict__ Ks,
                              const float* __restrict__ RTs,
                              unsigned* bbox, int V, int H, int W) {
  int total = V * H * W;
  float mnx = 3e38f, mny = 3e38f, mnz = 3e38f;
  float mxx = -3e38f, mxy = -3e38f, mxz = -3e38f;
  for (int p = blockIdx.x * blockDim.x + threadIdx.x; p < total;
       p += gridDim.x * blockDim.x) {
    int v = p / (H * W);
    int pix = p - v * H * W;
    int hy = pix / W;
    int ux = pix - hy * W;
    float d = depths[p];
    int m = masks[p];
    if (d > 0.f && m > 0) {
      const float* K = Ks + v * 9;
      const float* RT = RTs + v * 12;
      float camx = ((float)ux - K[2]) * d / K[0];
      float camy = ((float)hy - K[5]) * d / K[4];
      float camz = d;
      float px = camx - RT[3], py = camy - RT[7], pz = camz - RT[11];
      // world = R^T * (cam - T)
      float wx = RT[0] * px + RT[4] * py + RT[8] * pz;
      float wy = RT[1] * px + RT[5] * py + RT[9] * pz;
      float wz = RT[2] * px + RT[6] * py + RT[10] * pz;
      mnx = fminf(mnx, wx); mny = fminf(mny, wy); mnz = fminf(mnz, wz);
      mxx = fmaxf(mxx, wx); mxy = fmaxf(mxy, wy); mxz = fmaxf(mxz, wz);
    }
  }
  atomicMin(&bbox[0], encf(mnx));
  atomicMin(&bbox[1], encf(mny));
  atomicMin(&bbox[2], encf(mnz));
  atomicMax(&bbox[3], encf(mxx));
  atomicMax(&bbox[4], encf(mxy));
  atomicMax(&bbox[5], encf(mxz));
}

// ---------- pass 2: origin / vmax / vsz / trunc ----------
__global__ void k_bbox_finalize(const unsigned* bbox, float* params,
                                float* out_origin, float* out_bbox,
                                float* out_vsz, int res) {
  if (blockIdx.x != 0 || threadIdx.x != 0) return;
  float org[3], vmx[3];
  float span = -3e38f;
  for (int i = 0; i < 3; ++i) {
    float mn = decf(bbox[i]);
    float mx = decf(bbox[3 + i]);
    org[i] = mn - TH_BBOX;
    vmx[i] = mx + TH_BBOX;
    span = fmaxf(span, vmx[i] - org[i]);
  }
  float vsz = span / (float)res;
  float trunc = TRUNC_MULT * vsz;
  for (int i = 0; i < 3; ++i) {
    params[i] = org[i];
    params[3 + i] = vmx[i];
    out_origin[i] = org[i];
    out_bbox[i] = org[i];
    out_bbox[3 + i] = vmx[i];
  }
  params[6] = vsz;
  params[7] = trunc;
  *out_vsz = vsz;
}

// ---------- pass 3: TSDF integration (WMMA point transform) ----------
__global__ void __launch_bounds__(256)
k_fuse(const float* __restrict__ depths, const int* __restrict__ masks,
       const float* __restrict__ colors, const float* __restrict__ Ks,
       const float* __restrict__ RTs, const float* __restrict__ params,
       float* __restrict__ tsdf_out, float* __restrict__ col_out,
       float* __restrict__ occ_out, int res, int V, int H, int W) {
  int N = res * res * res;
  int tid = blockIdx.x * blockDim.x + threadIdx.x;
  int n = (tid < N) ? tid : (N - 1);  // clamp tail: keep EXEC all-ones for WMMA
  int lane = threadIdx.x & 31;
  bool lo = lane < 16;

  float ox = params[0], oy = params[1], oz = params[2];
  float vsz = params[6], trunc = params[7];
  float inv_trunc = 1.0f / trunc;     // hoist: replaces per-view IEEE divide

  int res2 = res * res;
  int gx = n / res2;
  int rem = n - gx * res2;
  int gy = rem / res;
  int gz = rem - gy * res;
  float wx = ox + ((float)gx + 0.5f) * vsz;
  float wy = oy + ((float)gy + 0.5f) * vsz;
  float wz = oz + ((float)gz + 0.5f) * vsz;

#ifdef USE_WMMA
  // B matrices (4x16, view-invariant): columns = homogeneous voxel coords.
  // Layout: v0 = {row K0 | row K2}, v1 = {row K1 | row K3}.
  int c1 = lane & 15, c2 = c1 + 16;
  float sx1 = __shfl(wx, c1, 32), sy1 = __shfl(wy, c1, 32), sz1 = __shfl(wz, c1, 32);
  float sx2 = __shfl(wx, c2, 32), sy2 = __shfl(wy, c2, 32), sz2 = __shfl(wz, c2, 32);
  v2f B1, B2;
  B1.x = lo ? sx1 : sz1;  B1.y = lo ? sy1 : 1.0f;
  B2.x = lo ? sx2 : sz2;  B2.y = lo ? sy2 : 1.0f;
  int M = lane & 15;
  int j = M & 7;                // A rows 0-2 and 8-10 carry [R|T] (duplicated)
  bool arow = (j < 3);
#endif

  float tsdf = 1.f, w = 0.f, col = 0.f;
  int HW = H * W;

  for (int v = 0; v < V; ++v) {
    const float* K = Ks + v * 9;
    const float* RT = RTs + v * 12;
    float fx = K[0], cxK = K[2], fy = K[4], cyK = K[5];
    float R00 = RT[0], R01 = RT[1], R02 = RT[2],  T0 = RT[3];
    float R10 = RT[4], R11 = RT[5], R12 = RT[6],  T1 = RT[7];
    float R20 = RT[8], R21 = RT[9], R22 = RT[10], T2 = RT[11];

    float camx, camy, camz;
#ifdef USE_WMMA
    float rj0 = (j == 0) ? R00 : ((j == 1) ? R10 : R20);
    float rj1 = (j == 0) ? R01 : ((j == 1) ? R11 : R21);
    float rj2 = (j == 0) ? R02 : ((j == 1) ? R12 : R22);
    float tj  = (j == 0) ? T0  : ((j == 1) ? T1  : T2 );
    v2f A;
    A.x = arow ? (lo ? rj0 : rj2) : 0.f;
    A.y = arow ? (lo ? rj1 : tj ) : 0.f;
    v8f zc = {};
    // D = [R|T] x [pts;1]  -> lane L<16 reads D1 rows 0..2 of its column,
    // lane 16+i reads D2 rows 8..10 (duplicated [R|T]) of column i.
    v8f D1 = __builtin_amdgcn_wmma_f32_16x16x4_f32(false, A, false, B1,
                                                   (short)0, zc, false, false);
    v8f D2 = __builtin_amdgcn_wmma_f32_16x16x4_f32(false, A, false, B2,
                                                   (short)0, zc, false, false);
    camx = lo ? D1[0] : D2[0];
    camy = lo ? D1[1] : D2[1];
    camz = lo ? D1[2] : D2[2];
#else
    camx = R00 * wx + R01 * wy + R02 * wz + T0;
    camy = R10 * wx + R11 * wy + R12 * wz + T1;
    camz = R20 * wx + R21 * wy + R22 * wz + T2;
#endif

    float zs = (camz == 0.f) ? 1e-9f : camz;
    float invz = 1.0f / zs;           // single divide, shared by u and v
    float up = fx * camx * invz + cxK;
    float vp = fy * camy * invz + cyK;
    int ui = (int)rintf(up);
    int vi = (int)rintf(vp);
    bool inb = (ui >= 0) && (ui < W) && (vi >= 0) && (vi < H) && (camz > 0.f);
    int uic = min(max(ui, 0), W - 1);
    int vic = min(max(vi, 0), H - 1);
    int pix2 = vic * W + uic;
    float d = depths[(size_t)v * HW + pix2];
    int m = masks[(size_t)v * HW + pix2];
    float sdf = d - camz;
    bool ok = inb && (d > 0.f) && (m > 0) && (sdf >= -trunc);
    float t_obs = fminf(1.f, sdf * inv_trunc);
    float w_obs = ok ? OBS_WEIGHT : 0.f;
    float w_new = w + w_obs;
    float invw = 1.f / fmaxf(w_new, 1e-9f);   // true divide: matches reference exactly
    float tsdf_n = (tsdf * w + t_obs * w_obs) * invw;
    tsdf = (w_new > 0.f) ? tsdf_n : tsdf;

    float r1v = colors[((size_t)v * 3 + 0) * HW + pix2];
    float g1v = colors[((size_t)v * 3 + 1) * HW + pix2];
    float b1v = colors[((size_t)v * 3 + 2) * HW + pix2];
    float c_obs = floorf(b1v * 65536.f + g1v * 256.f + r1v);
    float b0 = floorf(col * (1.f / 65536.f));
    float g0 = floorf((col - b0 * 65536.f) * (1.f / 256.f));
    float r0 = col - b0 * 65536.f - g0 * 256.f;
    float bb = floorf(c_obs * (1.f / 65536.f));
    float gg = floorf((c_obs - bb * 65536.f) * (1.f / 256.f));
    float rr2 = c_obs - bb * 65536.f - gg * 256.f;
    float rn = fminf(fmaxf((r0 * w + rr2 * w_obs) * invw, 0.f), 255.f);
    float gn = fminf(fmaxf((g0 * w + gg * w_obs) * invw, 0.f), 255.f);
    float bn = fminf(fmaxf((b0 * w + bb * w_obs) * invw, 0.f), 255.f);
    float ncol = floorf(bn) * 65536.f + floorf(gn) * 256.f + floorf(rn);
    col = (w_new > 0.f) ? ncol : col;
    w = w_new;
  }

  tsdf_out[n] = tsdf;
  col_out[n] = col;
  float th = TSDF_TH * vsz / trunc;
  occ_out[n] = (w > 0.f && tsdf > -th && tsdf < th) ? 1.f : 0.f;
}

// ---------- compaction: block partial sums ----------
#define SCAN_BT 256
#define SCAN_CHUNK 32
__global__ void k_partial_sums(const float* __restrict__ occ,
                               int* __restrict__ partials, int N) {
  __shared__ int s[SCAN_BT];
  int base = blockIdx.x * (SCAN_BT * SCAN_CHUNK) + threadIdx.x * SCAN_CHUNK;
  int sum = 0;
  for (int k = 0; k < SCAN_CHUNK; ++k) {
    int idx = base + k;
    if (idx < N) sum += (occ[idx] > 0.5f) ? 1 : 0;
  }
  s[threadIdx.x] = sum;
  __syncthreads();
  for (int off = SCAN_BT / 2; off > 0; off >>= 1) {
    if (threadIdx.x < off) s[threadIdx.x] += s[threadIdx.x + off];
    __syncthreads();
  }
  if (threadIdx.x == 0) partials[blockIdx.x] = s[0];
}

// ---------- compaction: serial exclusive scan of partials + count ----------
__global__ void k_serial_scan(int* partials, int P, float* num_out) {
  if (blockIdx.x != 0 || threadIdx.x != 0) return;
  int acc = 0;
  for (int i = 0; i < P; ++i) {
    int t = partials[i];
    partials[i] = acc;
    acc += t;
  }
  *num_out = (float)acc;
}

// ---------- 2x2x2 any-pool downsample ----------
__global__ void k_downsample(const float* __restrict__ occ_in,
                             float* __restrict__ occ_out, int r_out) {
  int N2 = r_out * r_out * r_out;
  int n = blockIdx.x * blockDim.x + threadIdx.x;
  if (n >= N2) return;
  int r_in = r_out * 2;
  int x = n / (r_out * r_out);
  int rem = n - x * r_out * r_out;
  int y = rem / r_out;
  int z = rem - y * r_out;
  float o = 0.f;
  for (int dx = 0; dx < 2; ++dx)
    for (int dy = 0; dy < 2; ++dy)
      for (int dz = 0; dz < 2; ++dz) {
        size_t idx = ((size_t)(2 * x + dx) * r_in + (2 * y + dy)) * r_in + (2 * z + dz);
        o = fmaxf(o, occ_in[idx]);
      }
  occ_out[n] = (o > 0.5f) ? 1.f : 0.f;
}

// ---------- compaction: rewrite occ (0/1) in place as idx / -1 ----------
__global__ void k_finalize_vol(float* __restrict__ vol,
                               const int* __restrict__ partials, int N) {
  __shared__ int s[SCAN_BT];
  int t = threadIdx.x;
  int base = blockIdx.x * (SCAN_BT * SCAN_CHUNK) + t * SCAN_CHUNK;
  int sum = 0;
  for (int k = 0; k < SCAN_CHUNK; ++k) {
    int idx = base + k;
    if (idx < N) sum += (vol[idx] > 0.5f) ? 1 : 0;
  }
  s[t] = sum;
  __syncthreads();
  if (t == 0) {
    int acc = 0;
    for (int i = 0; i < SCAN_BT; ++i) {
      int tmp = s[i];
      s[i] = acc;
      acc += tmp;
    }
  }
  __syncthreads();
  int pfx = partials[blockIdx.x] + s[t];
  for (int k = 0; k < SCAN_CHUNK; ++k) {
    int idx = base + k;
    if (idx < N) {
      bool o = vol[idx] > 0.5f;
      vol[idx] = o ? (float)pfx : -1.f;
      pfx += o ? 1 : 0;
    }
  }
}

extern "C" void kernel_launch(void* const* d_in, const int* in_sizes, int n_in,
                              void* d_out, int out_size, void* d_ws, size_t ws_size,
                              hipStream_t stream) {
  const float* colors = (const float*)d_in[0];
  const float* depths = (const float*)d_in[1];
  const int*   masks  = (const int*)d_in[2];
  const float* Ks     = (const float*)d_in[3];
  const float* RTs    = (const float*)d_in[4];

  int V = in_sizes[3] / 9;
  if (V <= 0) V = 1;
  long long HWll = (long long)in_sizes[1] / V;
  int W = 1;
  while ((long long)(W + 1) * (W + 1) <= HWll) ++W;  // assume square frames
  int H = W;

  // Recover res from out_size: 6 vol levels + num_occ(6) + tsdf + col + origin(3) + bbox(6) + vsz(1)
  long long res = 192;
  for (long long r = 32; r <= 1024; r += 32) {
    long long tot = 0, rr = r;
    for (int i = 0; i < 6; ++i) { tot += rr * rr * rr; rr /= 2; }
    tot += 2 * r * r * r + 16;
    if (tot == (long long)out_size) { res = r; break; }
  }
  long long N = res * res * res;

  long long volOff[6];
  long long off = 0, rr = res;
  for (int i = 0; i < 6; ++i) { volOff[i] = off; off += rr * rr * rr; rr /= 2; }
  long long numOff = off;    off += 6;
  long long tsdfOff = off;   off += N;
  long long colOff = off;    off += N;
  long long originOff = off; off += 3;
  long long bboxOff = off;   off += 6;
  long long vszOff = off;    off += 1;

  float* out = (float*)d_out;
  unsigned* bbox = (unsigned*)d_ws;                       // 6 u32
  float* params = (float*)((char*)d_ws + 64);             // 8 f32
  int* partials = (int*)((char*)d_ws + 256);              // <= ~1K ints

  k_init_ws<<<1, 32, 0, stream>>>(bbox);
  k_bbox_reduce<<<128, 256, 0, stream>>>(depths, masks, Ks, RTs, bbox, V, H, W);
  k_bbox_finalize<<<1, 1, 0, stream>>>(bbox, params, out + originOff,
                                       out + bboxOff, out + vszOff, (int)res);

  int fuseBlocks = (int)((N + 255) / 256);
  k_fuse<<<fuseBlocks, 256, 0, stream>>>(depths, masks, colors, Ks, RTs, params,
                                         out + tsdfOff, out + colOff,
                                         out + volOff[0], (int)res, V, H, W);

  long long r = res;
  for (int i = 0; i < 6; ++i) {
    long long Ni = r * r * r;
    int P = (int)((Ni + (SCAN_BT * SCAN_CHUNK) - 1) / (SCAN_BT * SCAN_CHUNK));
    k_partial_sums<<<P, SCAN_BT, 0, stream>>>(out + volOff[i], partials, (int)Ni);
    k_serial_scan<<<1, 1, 0, stream>>>(partials, P, out + numOff + i);
    if (i < 5) {
      long long r2 = r / 2;
      long long N2 = r2 * r2 * r2;
      k_downsample<<<(int)((N2 + 255) / 256), 256, 0, stream>>>(
          out + volOff[i], out + volOff[i + 1], (int)r2);
    }
    k_finalize_vol<<<P, SCAN_BT, 0, stream>>>(out + volOff[i], partials, (int)Ni);
    r /= 2;
  }
}